// DotAttention_47631187313068
// MI455X (gfx1250) — compile-verified
//
#include <hip/hip_runtime.h>
#include <hip/hip_bf16.h>

// Problem shape (from reference setup_inputs).
#define BB 8
#define TQ 2048
#define TK 2048
#define DD 1024
#define KC 32           // K-chunk staged in LDS per GEMM iteration

static __device__ __forceinline__ float negmask(int m) { return m ? 0.0f : -1e9f; }

typedef __attribute__((ext_vector_type(2))) float v2f;
typedef __attribute__((ext_vector_type(8))) float v8f;

// -----------------------------------------------------------------------------
// Kernel 1: scores[b, q, k] = <dec[b,q,:], enc[b,k,:]> + (1-mask[b,k]) * -1e9
// 128x64 output tile per 256-thread block; each wave owns 32x32 (4 WMMA accs).
// -----------------------------------------------------------------------------
__global__ __launch_bounds__(256) void scores_kernel(
    const float* __restrict__ dec, const float* __restrict__ enc,
    const int* __restrict__ mask, float* __restrict__ wts)
{
    const int bz  = blockIdx.z;
    const int qr0 = blockIdx.y * 128;  // query-row tile base
    const int kc0 = blockIdx.x * 64;   // key-col tile base

    const int t    = threadIdx.x;
    const int lane = t & 31;
    const int wave = t >> 5;
    const int lr   = lane & 15;        // lane row-in-half
    const int lh   = lane >> 4;        // lane half (K split for A/B frags)
    const int rw   = (wave & 3) * 32;  // wave row offset in tile (0..96)
    const int cw   = (wave >> 2) * 32; // wave col offset in tile (0,32)

    __shared__ float Qs[128 * 33];     // [row][k], stride 33 (conflict-free)
    __shared__ float Ks[64 * 33];      // [col][k]

    const float* decb = dec + ((size_t)bz * TQ + qr0) * DD;
    const float* encb = enc + ((size_t)bz * TK + kc0) * DD;

    v8f acc00 = {}, acc01 = {}, acc10 = {}, acc11 = {};

    const int qrow = t >> 1;           // 0..127 : Q tile row this thread loads
    const int qcol = (t & 1) * 16;     // 0,16   : 16-float chunk
    const int krow = t >> 2;           // 0..63  : K tile row
    const int kcol = (t & 3) * 8;      // 0,8,16,24

    for (int kt = 0; kt < DD; kt += KC) {
        // Stage Q tile (128 x 32) and K tile (64 x 32) into LDS.
        {
            const float* qsrc = decb + (size_t)qrow * DD + kt + qcol;
            float* qd = &Qs[qrow * 33 + qcol];
            #pragma unroll
            for (int c = 0; c < 4; ++c) {
                const float4 q = *(const float4*)(qsrc + 4 * c);
                qd[4 * c + 0] = q.x; qd[4 * c + 1] = q.y;
                qd[4 * c + 2] = q.z; qd[4 * c + 3] = q.w;
            }
            const float* ksrc = encb + (size_t)krow * DD + kt + kcol;
            float* kd = &Ks[krow * 33 + kcol];
            #pragma unroll
            for (int c = 0; c < 2; ++c) {
                const float4 k = *(const float4*)(ksrc + 4 * c);
                kd[4 * c + 0] = k.x; kd[4 * c + 1] = k.y;
                kd[4 * c + 2] = k.z; kd[4 * c + 3] = k.w;
            }
            if (kt + KC < DD) {        // prefetch next K-chunk (global_prefetch_b8)
                __builtin_prefetch(qsrc + KC, 0, 1);
                __builtin_prefetch(ksrc + KC, 0, 1);
            }
        }
        __syncthreads();

        #pragma unroll
        for (int kk = 0; kk < KC; kk += 4) {
            const int ko = kk + 2 * lh;
            v2f a0, a1, b0, b1;
            a0[0] = Qs[(rw + lr) * 33 + ko];
            a0[1] = Qs[(rw + lr) * 33 + ko + 1];
            a1[0] = Qs[(rw + 16 + lr) * 33 + ko];
            a1[1] = Qs[(rw + 16 + lr) * 33 + ko + 1];
            b0[0] = Ks[(cw + lr) * 33 + ko];
            b0[1] = Ks[(cw + lr) * 33 + ko + 1];
            b1[0] = Ks[(cw + 16 + lr) * 33 + ko];
            b1[1] = Ks[(cw + 16 + lr) * 33 + ko + 1];
            acc00 = __builtin_amdgcn_wmma_f32_16x16x4_f32(false, a0, false, b0,
                                                          (short)0, acc00, false, false);
            acc01 = __builtin_amdgcn_wmma_f32_16x16x4_f32(false, a0, false, b1,
                                                          (short)0, acc01, false, false);
            acc10 = __builtin_amdgcn_wmma_f32_16x16x4_f32(false, a1, false, b0,
                                                          (short)0, acc10, false, false);
            acc11 = __builtin_amdgcn_wmma_f32_16x16x4_f32(false, a1, false, b1,
                                                          (short)0, acc11, false, false);
        }
        __syncthreads();
    }

    // Additive mask + store. Col depends only on lane within each 16x16 tile.
    const int col0 = kc0 + cw + lr;
    const int col1 = col0 + 16;
    const float add0 = negmask(mask[bz * TK + col0]);
    const float add1 = negmask(mask[bz * TK + col1]);
    float* wb = wts + (size_t)bz * TQ * TK;
    #pragma unroll
    for (int r = 0; r < 8; ++r) {
        const int row0 = qr0 + rw + r + 8 * lh;
        const int row1 = row0 + 16;
        wb[(size_t)row0 * TK + col0] = acc00[r] + add0;
        wb[(size_t)row0 * TK + col1] = acc01[r] + add1;
        wb[(size_t)row1 * TK + col0] = acc10[r] + add0;
        wb[(size_t)row1 * TK + col1] = acc11[r] + add1;
    }
}

// -----------------------------------------------------------------------------
// Kernel 2: in-place row softmax over the last axis (Tk = 2048) of weights.
// One 256-thread block per (b, q) row; 8 elements per thread.
// -----------------------------------------------------------------------------
__global__ __launch_bounds__(256) void softmax_kernel(float* __restrict__ wts)
{
    const size_t row = blockIdx.x;
    float* w = wts + row * TK;
    const int t = threadIdx.x;

    float v[8];
    float mx = -3.402823466e38f;
    #pragma unroll
    for (int i = 0; i < 8; ++i) {
        v[i] = w[t + i * 256];
        mx = fmaxf(mx, v[i]);
    }

    __shared__ float red[256];
    red[t] = mx;
    __syncthreads();
    for (int s = 128; s > 0; s >>= 1) {
        if (t < s) red[t] = fmaxf(red[t], red[t + s]);
        __syncthreads();
    }
    mx = red[0];
    __syncthreads();

    float sum = 0.0f;
    #pragma unroll
    for (int i = 0; i < 8; ++i) {
        v[i] = expf(v[i] - mx);
        sum += v[i];
    }
    red[t] = sum;
    __syncthreads();
    for (int s = 128; s > 0; s >>= 1) {
        if (t < s) red[t] += red[t + s];
        __syncthreads();
    }
    const float inv = 1.0f / red[0];

    #pragma unroll
    for (int i = 0; i < 8; ++i)
        w[t + i * 256] = v[i] * inv;
}

// -----------------------------------------------------------------------------
// Kernel 3: context[b, q, d] = sum_k weights[b,q,k] * enc[b,k,d]
// 128x64 output tile; same 2x2 register-blocked WMMA scheme; B tile as [k][n].
// -----------------------------------------------------------------------------
__global__ __launch_bounds__(256) void context_kernel(
    const float* __restrict__ wts, const float* __restrict__ enc,
    float* __restrict__ ctx)
{
    const int bz  = blockIdx.z;
    const int qr0 = blockIdx.y * 128;  // query-row tile base
    const int dc0 = blockIdx.x * 64;   // d-col tile base

    const int t    = threadIdx.x;
    const int lane = t & 31;
    const int wave = t >> 5;
    const int lr   = lane & 15;
    const int lh   = lane >> 4;
    const int rw   = (wave & 3) * 32;
    const int cw   = (wave >> 2) * 32;

    __shared__ float Ws[128 * 33];     // A tile: [row][k]
    __shared__ float Vs[32 * 65];      // B tile: [k][n], stride 65

    const float* wb   = wts + (size_t)bz * TQ * TK + (size_t)qr0 * TK;
    const float* encb = enc + (size_t)bz * TK * DD + dc0;

    v8f acc00 = {}, acc01 = {}, acc10 = {}, acc11 = {};

    const int arow = t >> 1;           // 0..127 for Ws rows
    const int acol = (t & 1) * 16;     // k-chunk for Ws
    const int vrow = t >> 3;           // 0..31 for Vs k-rows
    const int vcol = (t & 7) * 8;      // n-chunk for Vs

    for (int kt = 0; kt < TK; kt += KC) {
        {
            const float* asrc = wb + (size_t)arow * TK + kt + acol;
            float* ad = &Ws[arow * 33 + acol];
            #pragma unroll
            for (int c = 0; c < 4; ++c) {
                const float4 a = *(const float4*)(asrc + 4 * c);
                ad[4 * c + 0] = a.x; ad[4 * c + 1] = a.y;
                ad[4 * c + 2] = a.z; ad[4 * c + 3] = a.w;
            }
            const float* vsrc = encb + (size_t)(kt + vrow) * DD + vcol;
            float* vd = &Vs[vrow * 65 + vcol];
            #pragma unroll
            for (int c = 0; c < 2; ++c) {
                const float4 b = *(const float4*)(vsrc + 4 * c);
                vd[4 * c + 0] = b.x; vd[4 * c + 1] = b.y;
                vd[4 * c + 2] = b.z; vd[4 * c + 3] = b.w;
            }
            if (kt + KC < TK) {
                __builtin_prefetch(asrc + KC, 0, 1);
                __builtin_prefetch(vsrc + (size_t)KC * DD, 0, 1);
            }
        }
        __syncthreads();

        #pragma unroll
        for (int kk = 0; kk < KC; kk += 4) {
            const int ko = kk + 2 * lh;
            v2f a0, a1, b0, b1;
            a0[0] = Ws[(rw + lr) * 33 + ko];
            a0[1] = Ws[(rw + lr) * 33 + ko + 1];
            a1[0] = Ws[(rw + 16 + lr) * 33 + ko];
            a1[1] = Ws[(rw + 16 + lr) * 33 + ko + 1];
            b0[0] = Vs[ko * 65 + cw + lr];
            b0[1] = Vs[(ko + 1) * 65 + cw + lr];
            b1[0] = Vs[ko * 65 + cw + 16 + lr];
            b1[1] = Vs[(ko + 1) * 65 + cw + 16 + lr];
            acc00 = __builtin_amdgcn_wmma_f32_16x16x4_f32(false, a0, false, b0,
                                                          (short)0, acc00, false, false);
            acc01 = __builtin_amdgcn_wmma_f32_16x16x4_f32(false, a0, false, b1,
                                                          (short)0, acc01, false, false);
            acc10 = __builtin_amdgcn_wmma_f32_16x16x4_f32(false, a1, false, b0,
                                                          (short)0, acc10, false, false);
            acc11 = __builtin_amdgcn_wmma_f32_16x16x4_f32(false, a1, false, b1,
                                                          (short)0, acc11, false, false);
        }
        __syncthreads();
    }

    float* cb = ctx + ((size_t)bz * TQ + qr0) * DD + dc0;
    #pragma unroll
    for (int r = 0; r < 8; ++r) {
        const int row0 = rw + r + 8 * lh;
        const int row1 = row0 + 16;
        cb[(size_t)row0 * DD + cw + lr]      = acc00[r];
        cb[(size_t)row0 * DD + cw + 16 + lr] = acc01[r];
        cb[(size_t)row1 * DD + cw + lr]      = acc10[r];
        cb[(size_t)row1 * DD + cw + 16 + lr] = acc11[r];
    }
}

// -----------------------------------------------------------------------------
// Host entry
// -----------------------------------------------------------------------------
extern "C" void kernel_launch(void* const* d_in, const int* in_sizes, int n_in,
                              void* d_out, int out_size, void* d_ws, size_t ws_size,
                              hipStream_t stream)
{
    const float* dec  = (const float*)d_in[0];   // [B, Tq, D]
    const float* enc  = (const float*)d_in[1];   // [B, Tk, D]
    const int*   mask = (const int*)d_in[2];     // [B, Tk]

    float* ctx = (float*)d_out;                              // [B, Tq, D]
    float* wts = (float*)d_out + (size_t)BB * TQ * DD;       // [B, Tq, Tk]

    scores_kernel<<<dim3(TK / 64, TQ / 128, BB), 256, 0, stream>>>(dec, enc, mask, wts);
    softmax_kernel<<<dim3(BB * TQ), 256, 0, stream>>>(wts);
    context_kernel<<<dim3(DD / 64, TQ / 128, BB), 256, 0, stream>>>(wts, enc, ctx);
}